// GaussianAdapter_5617817223353
// MI455X (gfx1250) — compile-verified
//
#include <hip/hip_runtime.h>
#include <stdint.h>

typedef unsigned int u32;
typedef __attribute__((ext_vector_type(4))) u32 v4u;
typedef __attribute__((ext_vector_type(8))) u32 v8u;

#define TILE 128           // gaussians per block == blockDim.x (R % TILE == 0)
#define D_IN 82            // 7 + 3*25 raw floats per gaussian
#define D_SH 25
#define SH_F (3 * D_SH)    // 75 sh floats per gaussian

// 0.1 * 0.25^degree mask per SH coefficient
__constant__ float c_mask[D_SH] = {
    1.0f,
    0.025f, 0.025f, 0.025f,
    0.00625f, 0.00625f, 0.00625f, 0.00625f, 0.00625f,
    0.0015625f, 0.0015625f, 0.0015625f, 0.0015625f, 0.0015625f, 0.0015625f, 0.0015625f,
    0.000390625f, 0.000390625f, 0.000390625f, 0.000390625f, 0.000390625f,
    0.000390625f, 0.000390625f, 0.000390625f, 0.000390625f};

// ----------------------------------------------------------------------
// Issue one TDM transfer (1 row of n_el 4-byte elements, contiguous both
// sides). D# per CDNA5 ISA ch.8: group0 (4 SGPRs) + group1 (8 SGPRs).
// Direction chosen by opcode; descriptors identical for load and store.
// ----------------------------------------------------------------------
__device__ __forceinline__ void tdm_copy_1d(u32 lds_off, uint64_t gaddr, u32 n_el,
                                            bool store) {
  v4u d0;
  d0[0] = 1u;                                                // count=1, user desc
  d0[1] = lds_off;                                           // lds_addr (bytes)
  d0[2] = (u32)gaddr;                                        // global_addr[31:0]
  d0[3] = (u32)((gaddr >> 32) & 0x1FFFFFFull) | (2u << 30);  // addr[56:32] | type=2

  v8u d1;
  d1[0] = (2u << 16);                  // workgroup_mask=0, data_size=4B
  d1[1] = (n_el & 0xFFFFu) << 16;      // tensor_dim0[15:0]
  d1[2] = (n_el >> 16) | (1u << 16);   // tensor_dim0[31:16], tensor_dim1=1 (lo)
  d1[3] = (n_el & 0xFFFFu) << 16;      // tensor_dim1 hi=0, tile_dim0=n_el
  d1[4] = 1u;                          // tile_dim1=1, tile_dim2=0
  d1[5] = n_el;                        // tensor_dim0_stride lo
  d1[6] = 0u;                          // stride0 hi, stride1 lo
  d1[7] = 0u;                          // stride1 hi

  if (store)
    asm volatile("tensor_store_from_lds %0, %1" ::"s"(d0), "s"(d1) : "memory");
  else
    asm volatile("tensor_load_to_lds %0, %1" ::"s"(d0), "s"(d1) : "memory");
}

__global__ __launch_bounds__(TILE) void gaussian_adapter_kernel(
    const float* __restrict__ ext, const float* __restrict__ K,
    const float* __restrict__ coords, const float* __restrict__ depths,
    const float* __restrict__ opac, const float* __restrict__ raw,
    const int* __restrict__ hp, const int* __restrict__ wp,
    float* __restrict__ out, int N, int R) {
  __shared__ __align__(16) float lds_raw[TILE * D_IN];  // 41,984 B

  const int tid = threadIdx.x;
  const int view = blockIdx.y;                  // no division needed
  const int g0 = view * R + blockIdx.x * TILE;  // first gaussian of this tile
  const int g = g0 + tid;
  const u32 lds_base = (u32)(uintptr_t)(&lds_raw[0]);

  // ---- kick off the TDM load of the raw tile (TILE x 82 f32, contiguous)
  if (tid < 32) {  // wave 0 only (uniform branch); TDM ignores EXEC
    tdm_copy_1d(lds_base, (uint64_t)(uintptr_t)(raw + (size_t)g0 * D_IN),
                (u32)(TILE * D_IN), /*store=*/false);
  }

  // ---- overlap DMA with per-view constants + coalesced per-lane loads
  const float eps = 1e-8f;
  const float* Kv = K + view * 9;
  const float ka = Kv[0], kb = Kv[1], kc = Kv[2];
  const float kd = Kv[3], ke = Kv[4], kf = Kv[5];
  const float kg = Kv[6], kh = Kv[7], ki = Kv[8];
  // adjugate of K (for K^-1 = adj/det)
  const float A00 = ke * ki - kf * kh, A01 = kc * kh - kb * ki, A02 = kb * kf - kc * ke;
  const float A10 = kf * kg - kd * ki, A11 = ka * ki - kc * kg, A12 = kc * kd - ka * kf;
  const float A20 = kd * kh - ke * kg, A21 = kb * kg - ka * kh, A22 = ka * ke - kb * kd;
  const float idet = 1.0f / (ka * A00 + kb * A10 + kc * A20);

  const int h = *hp, w = *wp;
  const float pw = 1.0f / (float)w, ph = 1.0f / (float)h;
  const float det2 = ka * ke - kb * kd;
  const float multiplier = 0.1f * ((ke * pw - kb * ph) + (-kd * pw + ka * ph)) / det2;

  const float* E = ext + view * 16;  // row-major 4x4 c2w
  const float C00 = E[0], C01 = E[1], C02 = E[2], t0 = E[3];
  const float C10 = E[4], C11 = E[5], C12 = E[6], t1 = E[7];
  const float C20 = E[8], C21 = E[9], C22 = E[10], t2 = E[11];

  const float depth = depths[g];
  const float cu = coords[2 * (size_t)g], cv = coords[2 * (size_t)g + 1];
  const float op_v = opac[g];

  // ---- ray dirs -> means (independent of LDS)
  float dx = (A00 * cu + A01 * cv + A02) * idet;
  float dy = (A10 * cu + A11 * cv + A12) * idet;
  float dz = (A20 * cu + A21 * cv + A22) * idet;
  const float rn = rsqrtf(dx * dx + dy * dy + dz * dz);
  dx *= rn; dy *= rn; dz *= rn;
  const float wxd = C00 * dx + C01 * dy + C02 * dz;
  const float wyd = C10 * dx + C11 * dy + C12 * dz;
  const float wzd = C20 * dx + C21 * dy + C22 * dz;
  const float mx = t0 + wxd * depth, my = t1 + wyd * depth, mz = t2 + wzd * depth;

  // ---- wait for the tile (no-op for waves with TENSORcnt==0), publish LDS
#if __has_builtin(__builtin_amdgcn_s_wait_tensorcnt)
  __builtin_amdgcn_s_wait_tensorcnt(0);
#else
  asm volatile("s_wait_tensorcnt 0x0" ::: "memory");
#endif
  __syncthreads();

  // ---- per-gaussian math (reads first 7 floats of its LDS row)
  const float* rg = &lds_raw[tid * D_IN];

  const float mul = depth * multiplier;
  const float sx = (0.5f + 14.5f / (1.0f + __expf(-rg[0]))) * mul;
  const float sy = (0.5f + 14.5f / (1.0f + __expf(-rg[1]))) * mul;
  const float sz = (0.5f + 14.5f / (1.0f + __expf(-rg[2]))) * mul;

  float qr = rg[3], qi = rg[4], qj = rg[5], qk = rg[6];
  const float qn = sqrtf(qr * qr + qi * qi + qj * qj + qk * qk) + eps;
  qr /= qn; qi /= qn; qj /= qn; qk /= qn;
  const float two_s = 2.0f / (qr * qr + qi * qi + qj * qj + qk * qk);

  const float R00 = 1.0f - two_s * (qj * qj + qk * qk);
  const float R01 = two_s * (qi * qj - qk * qr);
  const float R02 = two_s * (qi * qk + qj * qr);
  const float R10 = two_s * (qi * qj + qk * qr);
  const float R11 = 1.0f - two_s * (qi * qi + qk * qk);
  const float R12 = two_s * (qj * qk - qi * qr);
  const float R20 = two_s * (qi * qk - qj * qr);
  const float R21 = two_s * (qj * qk + qi * qr);
  const float R22 = 1.0f - two_s * (qi * qi + qj * qj);

  const float s2x = sx * sx, s2y = sy * sy, s2z = sz * sz;
  // M = R diag(s2) R^T (symmetric)
  const float M00 = R00 * R00 * s2x + R01 * R01 * s2y + R02 * R02 * s2z;
  const float M01 = R00 * R10 * s2x + R01 * R11 * s2y + R02 * R12 * s2z;
  const float M02 = R00 * R20 * s2x + R01 * R21 * s2y + R02 * R22 * s2z;
  const float M11 = R10 * R10 * s2x + R11 * R11 * s2y + R12 * R12 * s2z;
  const float M12 = R10 * R20 * s2x + R11 * R21 * s2y + R12 * R22 * s2z;
  const float M22 = R20 * R20 * s2x + R21 * R21 * s2y + R22 * R22 * s2z;
  // T = C * M
  const float T00 = C00 * M00 + C01 * M01 + C02 * M02;
  const float T01 = C00 * M01 + C01 * M11 + C02 * M12;
  const float T02 = C00 * M02 + C01 * M12 + C02 * M22;
  const float T10 = C10 * M00 + C11 * M01 + C12 * M02;
  const float T11 = C10 * M01 + C11 * M11 + C12 * M12;
  const float T12 = C10 * M02 + C11 * M12 + C12 * M22;
  const float T20 = C20 * M00 + C21 * M01 + C22 * M02;
  const float T21 = C20 * M01 + C21 * M11 + C22 * M12;
  const float T22 = C20 * M02 + C21 * M12 + C22 * M22;
  // cov = T * C^T
  const float V00 = T00 * C00 + T01 * C01 + T02 * C02;
  const float V01 = T00 * C10 + T01 * C11 + T02 * C12;
  const float V02 = T00 * C20 + T01 * C21 + T02 * C22;
  const float V10 = T10 * C00 + T11 * C01 + T12 * C02;
  const float V11 = T10 * C10 + T11 * C11 + T12 * C12;
  const float V12 = T10 * C20 + T11 * C21 + T12 * C22;
  const float V20 = T20 * C00 + T21 * C01 + T22 * C02;
  const float V21 = T20 * C10 + T21 * C11 + T22 * C12;
  const float V22 = T20 * C20 + T21 * C21 + T22 * C22;

  // ---- output layout (flat concat): means 3N | cov 9N | scales 3N | rot 4N
  //                                   | sh 75N | opacity N
  const size_t sN = (size_t)N;
  float* o_means = out;
  float* o_cov = out + 3 * sN;
  float* o_scales = out + 12 * sN;
  float* o_rot = out + 15 * sN;
  float* o_sh = out + 19 * sN;
  float* o_op = out + 94 * sN;

  // contiguous-per-lane stores
  *(float4*)(o_rot + 4 * (size_t)g) = make_float4(qr, qi, qj, qk);
  o_op[g] = op_v;

  // SH: mask-scaled copy; block's output span is contiguous -> float4 stores
  {
    float4* dst = (float4*)(o_sh + (size_t)g0 * SH_F);
    const int nvec = TILE * SH_F / 4;  // 2400
    for (int v4i = tid; v4i < nvec; v4i += TILE) {
      float4 o4;
      float* o4p = (float*)&o4;
#pragma unroll
      for (int c = 0; c < 4; ++c) {
        const int e = v4i * 4 + c;
        const int r = e / SH_F;
        const int j = e - r * SH_F;
        o4p[c] = lds_raw[r * D_IN + 7 + j] * c_mask[j % D_SH];
      }
      dst[v4i] = o4;
    }
  }
  __syncthreads();  // done reading raw tile; reuse LDS as store-staging

  // stage cov/means/scales contiguously, then DMA them out with the TDM
  float* sc_cov = lds_raw;                // TILE*9 floats
  float* sc_mean = lds_raw + TILE * 9;    // TILE*3 floats
  float* sc_scale = lds_raw + TILE * 12;  // TILE*3 floats
  sc_cov[tid * 9 + 0] = V00; sc_cov[tid * 9 + 1] = V01; sc_cov[tid * 9 + 2] = V02;
  sc_cov[tid * 9 + 3] = V10; sc_cov[tid * 9 + 4] = V11; sc_cov[tid * 9 + 5] = V12;
  sc_cov[tid * 9 + 6] = V20; sc_cov[tid * 9 + 7] = V21; sc_cov[tid * 9 + 8] = V22;
  sc_mean[tid * 3 + 0] = mx; sc_mean[tid * 3 + 1] = my; sc_mean[tid * 3 + 2] = mz;
  sc_scale[tid * 3 + 0] = sx; sc_scale[tid * 3 + 1] = sy; sc_scale[tid * 3 + 2] = sz;
  __syncthreads();

  if (tid < 32) {
    tdm_copy_1d(lds_base, (uint64_t)(uintptr_t)(o_cov + (size_t)g0 * 9),
                (u32)(TILE * 9), /*store=*/true);
    tdm_copy_1d(lds_base + TILE * 9 * 4, (uint64_t)(uintptr_t)(o_means + (size_t)g0 * 3),
                (u32)(TILE * 3), /*store=*/true);
    tdm_copy_1d(lds_base + TILE * 12 * 4, (uint64_t)(uintptr_t)(o_scales + (size_t)g0 * 3),
                (u32)(TILE * 3), /*store=*/true);
    // no explicit wait: S_ENDPGM performs an implicit wait-idle (TENSORcnt==0),
    // and LDS stays allocated until the workgroup retires.
  }
}

extern "C" void kernel_launch(void* const* d_in, const int* in_sizes, int n_in,
                              void* d_out, int out_size, void* d_ws, size_t ws_size,
                              hipStream_t stream) {
  (void)n_in; (void)out_size; (void)d_ws; (void)ws_size;
  const float* ext = (const float*)d_in[0];
  const float* K = (const float*)d_in[1];
  const float* coords = (const float*)d_in[2];
  const float* depths = (const float*)d_in[3];
  const float* opac = (const float*)d_in[4];
  const float* raw = (const float*)d_in[5];
  const int* hp = (const int*)d_in[6];
  const int* wp = (const int*)d_in[7];

  const int N = in_sizes[3];            // B*V*R gaussians
  const int n_views = in_sizes[0] / 16; // B*V
  const int R = N / n_views;            // rays per view (multiple of TILE)

  dim3 grid(R / TILE, n_views, 1);
  hipLaunchKernelGGL(gaussian_adapter_kernel, grid, dim3(TILE), 0, stream,
                     ext, K, coords, depths, opac, raw, hp, wp, (float*)d_out, N, R);
}